// GLIPHead_65833258713485
// MI455X (gfx1250) — compile-verified
//
#include <hip/hip_runtime.h>

// ---------------------------------------------------------------------------
// GLIP detection head for MI455X (gfx1250, wave32).
// Dense linear algebra on the WMMA f32 16x16x4 matrix pipe; conv im2col is
// staged through LDS via async global->LDS loads (ASYNCcnt path).
// ---------------------------------------------------------------------------

typedef float v2f __attribute__((ext_vector_type(2)));
typedef float v8f __attribute__((ext_vector_type(8)));

#define NEGV  -1000000000.0f
#define IMGSZ 832.0f
#define LOGC  4.135166556742356f

#if defined(__has_builtin)
#if __has_builtin(__builtin_amdgcn_global_load_async_to_lds_b32) && \
    __has_builtin(__builtin_amdgcn_s_wait_asynccnt)
#define HAVE_ASYNC 1
#endif
#endif
#ifndef HAVE_ASYNC
#define HAVE_ASYNC 0
#endif

typedef __attribute__((address_space(1))) int* gptr_t;
typedef __attribute__((address_space(3))) int* lptr_t;

__device__ __forceinline__ v8f wmma4(v2f a, v2f b, v8f c) {
  // D = A(16x4) * B(4x16) + C(16x16), f32, wave32
  return __builtin_amdgcn_wmma_f32_16x16x4_f32(false, a, false, b,
                                               (short)0, c, false, false);
}

__device__ __forceinline__ float clip832(float v) {
  return fminf(fmaxf(v, 0.0f), IMGSZ);
}

// --------------------------- bias init for fused 1x1 convs -----------------
__global__ void k_init_bias(float* __restrict__ obj, float* __restrict__ dlt,
                            const float* __restrict__ bc, const float* __restrict__ bb,
                            int HW) {
  int i = blockIdx.x * blockDim.x + threadIdx.x;
  int b = blockIdx.z;
  if (i < 3 * HW) obj[(size_t)b * 3 * HW + i] = bc[i / HW];
  int j = i - 3 * HW;
  if (j >= 0 && j < 12 * HW) dlt[(size_t)b * 12 * HW + j] = bb[j / HW];
}

// --------------------------- 3x3 conv + ReLU + fused 1x1 cls/bbox ----------
// Workgroup = 8 waves: 128 outC x 16 pixels. im2col patch staged in LDS per
// 72-wide K chunk (8 cin x 9 taps); 18 WMMA per chunk, fully unrolled.
#define KC  72
#define KCP 76  // padded row stride (floats): 16 lanes hit distinct banks

__global__ __launch_bounds__(256)
void k_conv_rpn(const float* __restrict__ feat,   // (B,256,H,W)
                const float* __restrict__ wc,     // (256,256,3,3)
                const float* __restrict__ bc,     // (256)
                const float* __restrict__ wcls,   // (3,256)
                const float* __restrict__ wbbox,  // (12,256)
                float* __restrict__ obj,          // (B,3,H,W)  pre-filled with bias
                float* __restrict__ dlt,          // (B,12,H,W) pre-filled with bias
                int H, int W) {
  const int t = threadIdx.x;
  const int lane = t & 31;
  const int wave = t >> 5;
  const int HW = H * W;
  const int oc_base = (blockIdx.y * 8 + wave) * 16;
  const int b = blockIdx.z;
  const int tile0 = blockIdx.x * 16;

  __shared__ float lB[16][KCP];  // [pixel][k-in-chunk]

  const float* fb = feat + (size_t)b * 256 * HW;
  const int m = lane & 15;
  const int n = lane & 15;
  const int koff = (lane >> 4) * 2;
  const float* wrow = wc + (size_t)(oc_base + m) * 2304;

  v8f acc = {};
  for (int k0 = 0; k0 < 2304; k0 += KC) {
    // ---- cooperative im2col fill: lB[n][kk] = patch(px=tile0+n, ki=k0+kk)
    for (int e = t; e < KC * 16; e += 256) {
      int kk = e >> 4;  // 0..71
      int nn = e & 15;
      int ki = k0 + kk;
      int cin = ki / 9;
      int rem = ki - cin * 9;
      int ky = rem / 3, kx = rem - ky * 3;
      int px = tile0 + nn;
      int y = px / W, x = px - y * W;
      int yy = y + ky - 1, xx = x + kx - 1;
      bool ok = (px < HW) && yy >= 0 && yy < H && xx >= 0 && xx < W;
#if HAVE_ASYNC
      if (ok) {
        const float* gp = fb + (size_t)cin * HW + (size_t)yy * W + xx;
        __builtin_amdgcn_global_load_async_to_lds_b32(
            (gptr_t)(size_t)gp, (lptr_t)&lB[nn][kk], 0, 0);
      } else {
        lB[nn][kk] = 0.0f;
      }
#else
      lB[nn][kk] = ok ? fb[(size_t)cin * HW + (size_t)yy * W + xx] : 0.0f;
#endif
    }
#if HAVE_ASYNC
    __builtin_amdgcn_s_wait_asynccnt(0);
#endif
    __syncthreads();

    if (k0 + KC < 2304) __builtin_prefetch(wrow + k0 + KC, 0, 0);
    // ---- 18 back-to-back WMMA on the staged chunk
#pragma unroll
    for (int kk = 0; kk < KC; kk += 4) {
      v2f a, bv;
      a.x = wrow[k0 + kk + koff];
      a.y = wrow[k0 + kk + koff + 1];
      bv.x = lB[n][kk + koff];
      bv.y = lB[n][kk + koff + 1];
      acc = wmma4(a, bv, acc);
    }
    __syncthreads();
  }

  const int px = tile0 + n;
  if (px >= HW) return;

  float hidv[8];
#pragma unroll
  for (int r = 0; r < 8; ++r) {
    int mm = oc_base + ((lane < 16) ? r : r + 8);
    float v = acc[r] + bc[mm];
    hidv[r] = v > 0.0f ? v : 0.0f;
  }
  // fused 1x1 convs: accumulate this 16-channel slab's contribution
#pragma unroll 1
  for (int oc = 0; oc < 15; ++oc) {
    const float* w1 = (oc < 3) ? (wcls + oc * 256) : (wbbox + (oc - 3) * 256);
    float p = 0.0f;
#pragma unroll
    for (int r = 0; r < 8; ++r) {
      int mm = oc_base + ((lane < 16) ? r : r + 8);
      p += hidv[r] * w1[mm];
    }
    float* outp = (oc < 3) ? (obj + ((size_t)b * 3 + oc) * HW + px)
                           : (dlt + ((size_t)b * 12 + (oc - 3)) * HW + px);
    atomicAdd(outp, p);
  }
}

// --------------------------- anchor decode -----------------------------------
__global__ void k_decode(const float* __restrict__ obj, const float* __restrict__ dlt,
                         float* __restrict__ boxes, float* __restrict__ scores,
                         int H, int W, float stride, float size) {
  int i = blockIdx.x * blockDim.x + threadIdx.x;
  int b = blockIdx.z;
  int HW = H * W;
  if (i >= 3 * HW) return;
  int p = i / 3, a = i - p * 3;
  int y = p / W, x = p - y * W;
  const float RT[3] = {0.5f, 1.0f, 2.0f};
  float sq = sqrtf(RT[a]);
  float h = size * sq, w = size / sq;
  float cx = x * stride, cy = y * stride;
  const float* D = dlt + (size_t)b * 12 * HW;
  float d0 = D[(a * 4 + 0) * HW + p];
  float d1 = D[(a * 4 + 1) * HW + p];
  float dw = fminf(D[(a * 4 + 2) * HW + p], LOGC);
  float dh = fminf(D[(a * 4 + 3) * HW + p], LOGC);
  float pcx = d0 * w + cx, pcy = d1 * h + cy;
  float pw = expf(dw) * w, ph = expf(dh) * h;
  size_t o = (size_t)b * 3 * HW + i;
  boxes[o * 4 + 0] = clip832(pcx - 0.5f * pw);
  boxes[o * 4 + 1] = clip832(pcy - 0.5f * ph);
  boxes[o * 4 + 2] = clip832(pcx + 0.5f * pw);
  boxes[o * 4 + 3] = clip832(pcy + 0.5f * ph);
  scores[o] = obj[((size_t)b * 3 + a) * HW + p];
}

// --------------------------- per-level top-k ---------------------------------
__global__ __launch_bounds__(1024)
void k_topk(const float* __restrict__ boxes, const float* __restrict__ scores,
            float* __restrict__ scr, float* __restrict__ coff,
            float* __restrict__ craw, float* __restrict__ csc,
            int count, int k, float lvl, int cbase, int cstride) {
  int b = blockIdx.x, t = threadIdx.x;
  const float* Bx = boxes + (size_t)b * count * 4;
  const float* S0 = scores + (size_t)b * count;
  float* S = scr + (size_t)b * count;
  __shared__ float sv[1024];
  __shared__ int si[1024];
  for (int i = t; i < count; i += 1024) S[i] = S0[i];
  __syncthreads();
  for (int it = 0; it < k; ++it) {
    float best = -3.4e38f; int bi = 0;
    for (int i = t; i < count; i += 1024) { float v = S[i]; if (v > best) { best = v; bi = i; } }
    sv[t] = best; si[t] = bi; __syncthreads();
    for (int st = 512; st > 0; st >>= 1) {
      if (t < st && sv[t + st] > sv[t]) { sv[t] = sv[t + st]; si[t] = si[t + st]; }
      __syncthreads();
    }
    if (t == 0) {
      int j = si[0]; float sc = sv[0];
      float x1 = Bx[j * 4], y1 = Bx[j * 4 + 1], x2 = Bx[j * 4 + 2], y2 = Bx[j * 4 + 3];
      size_t o = (size_t)b * cstride + cbase + it;
      craw[o * 4 + 0] = x1; craw[o * 4 + 1] = y1; craw[o * 4 + 2] = x2; craw[o * 4 + 3] = y2;
      float add = lvl * 10000.0f;
      coff[o * 4 + 0] = x1 + add; coff[o * 4 + 1] = y1 + add;
      coff[o * 4 + 2] = x2 + add; coff[o * 4 + 3] = y2 + add;
      bool ok = (x2 - x1 >= 1e-3f) && (y2 - y1 >= 1e-3f);
      csc[o] = ok ? sc : NEGV;
      S[j] = -3.4e38f;
    }
    __syncthreads();
  }
}

// --------------------------- greedy NMS --------------------------------------
__global__ __launch_bounds__(1024)
void k_nms(const float* __restrict__ nb, const float* __restrict__ sin,
           float* __restrict__ scr, int count, int k, float thresh,
           int* __restrict__ oidx, float* __restrict__ oval, float* __restrict__ osc) {
  int b = blockIdx.x, t = threadIdx.x;
  const float* B_ = nb + (size_t)b * count * 4;
  const float* SI = sin + (size_t)b * count;
  float* S = scr + (size_t)b * count;
  __shared__ float sv[1024];
  __shared__ int si[1024];
  __shared__ float bx[4];
  __shared__ int bj;
  __shared__ float bsc;
  for (int i = t; i < count; i += 1024) S[i] = SI[i];
  __syncthreads();
  for (int it = 0; it < k; ++it) {
    float best = -3.4e38f; int bi = 0;
    for (int i = t; i < count; i += 1024) { float v = S[i]; if (v > best) { best = v; bi = i; } }
    sv[t] = best; si[t] = bi; __syncthreads();
    for (int st = 512; st > 0; st >>= 1) {
      if (t < st && sv[t + st] > sv[t]) { sv[t] = sv[t + st]; si[t] = si[t + st]; }
      __syncthreads();
    }
    if (t == 0) {
      bj = si[0]; bsc = sv[0];
      bx[0] = B_[bj * 4]; bx[1] = B_[bj * 4 + 1];
      bx[2] = B_[bj * 4 + 2]; bx[3] = B_[bj * 4 + 3];
    }
    __syncthreads();
    float X1 = bx[0], Y1 = bx[1], X2 = bx[2], Y2 = bx[3];
    float a = fmaxf(X2 - X1, 0.0f) * fmaxf(Y2 - Y1, 0.0f);
    for (int i = t; i < count; i += 1024) {
      float x1 = fmaxf(X1, B_[i * 4]),     y1 = fmaxf(Y1, B_[i * 4 + 1]);
      float x2 = fminf(X2, B_[i * 4 + 2]), y2 = fminf(Y2, B_[i * 4 + 3]);
      float inter = fmaxf(x2 - x1, 0.0f) * fmaxf(y2 - y1, 0.0f);
      float bb = fmaxf(B_[i * 4 + 2] - B_[i * 4], 0.0f) *
                 fmaxf(B_[i * 4 + 3] - B_[i * 4 + 1], 0.0f);
      float iou = inter / (a + bb - inter + 1e-6f);
      if (iou > thresh) S[i] = NEGV;
    }
    if (t == 0) {
      S[bj] = NEGV;
      oidx[b * k + it] = bj;
      oval[b * k + it] = (bsc > NEGV * 0.5f) ? 1.0f : 0.0f;
      osc[b * k + it] = bsc;
    }
    __syncthreads();
  }
}

// --------------------------- proposal gather ---------------------------------
__global__ void k_props(const float* __restrict__ craw, const int* __restrict__ idx,
                        const float* __restrict__ val, float* __restrict__ props,
                        int cstride) {
  int i = blockIdx.x * blockDim.x + threadIdx.x;
  int b = blockIdx.z;
  if (i >= 1000) return;
  int j = idx[b * 1000 + i];
  float v = val[b * 1000 + i];
  for (int c = 0; c < 4; ++c)
    props[((size_t)b * 1000 + i) * 4 + c] = craw[((size_t)b * cstride + j) * 4 + c] * v;
}

// --------------------------- multi-scale ROI align ---------------------------
__global__ __launch_bounds__(256)
void k_msroi(const float* __restrict__ f0, const float* __restrict__ f1,
             const float* __restrict__ f2, const float* __restrict__ f3,
             const float* __restrict__ rois, size_t roiImgStride,
             float* __restrict__ outp, size_t outImgStride) {
  int blk = blockIdx.x;
  int roi = blk / 49;
  int cell = blk - roi * 49;
  int oy = cell / 7, ox = cell - oy * 7;
  int b = blockIdx.z;
  int c = threadIdx.x;  // 0..255
  const float* R = rois + (size_t)b * roiImgStride + (size_t)roi * 4;
  float x1 = R[0], y1 = R[1], x2 = R[2], y2 = R[3];
  float w = fmaxf(x2 - x1, 0.0f), h = fmaxf(y2 - y1, 0.0f);
  float lvlf = floorf(4.0f + log2f(sqrtf(w * h) / 224.0f + 1e-8f));
  int li = (int)(fminf(fmaxf(lvlf, 2.0f), 5.0f)) - 2;
  const float* feats[4] = {f0, f1, f2, f3};
  const int HS[4] = {208, 104, 52, 26};
  const float SCL[4] = {0.25f, 0.125f, 0.0625f, 0.03125f};
  int M = HS[li];
  float scale = SCL[li];
  const float* F = feats[li] + ((size_t)b * 256 + c) * M * M;
  float bx1 = x1 * scale, by1 = y1 * scale;
  float rw = fmaxf(x2 * scale - bx1, 1.0f);
  float rh = fmaxf(y2 * scale - by1, 1.0f);
  float acc = 0.0f;
#pragma unroll
  for (int sy = 0; sy < 2; ++sy)
#pragma unroll
    for (int sx = 0; sx < 2; ++sx) {
      float ty = (oy * 2 + sy + 0.5f) / 14.0f;
      float tx = (ox * 2 + sx + 0.5f) / 14.0f;
      float yv = fminf(fmaxf(by1 + ty * rh, 0.0f), (float)(M - 1));
      float xv = fminf(fmaxf(bx1 + tx * rw, 0.0f), (float)(M - 1));
      int y0 = (int)floorf(yv); int y1i = y0 + 1; if (y1i > M - 1) y1i = M - 1;
      int x0 = (int)floorf(xv); int x1i = x0 + 1; if (x1i > M - 1) x1i = M - 1;
      float fy = yv - y0, fx = xv - x0;
      acc += F[y0 * M + x0] * (1 - fy) * (1 - fx) + F[y0 * M + x1i] * (1 - fy) * fx +
             F[y1i * M + x0] * fy * (1 - fx) + F[y1i * M + x1i] * fy * fx;
    }
  outp[(size_t)b * outImgStride + (size_t)roi * 12544 + c * 49 + cell] = acc * 0.25f;
}

// --------------------------- generic WMMA GEMM -------------------------------
// C[M,N] = act(A[M,K] x B + bias); B is (K,N) or, if transB, (N,K).
__global__ __launch_bounds__(256)
void k_gemm(const float* __restrict__ A, const float* __restrict__ Bm,
            const float* __restrict__ bias, float* __restrict__ C,
            int M, int N, int K, int lda, int ldb, int ldc,
            size_t sA, size_t sB, size_t sC, int transB, int relu) {
  int lane = threadIdx.x & 31, wave = threadIdx.x >> 5;
  int z = blockIdx.z;
  int m0 = blockIdx.x * 16;
  int n0 = (blockIdx.y * 8 + wave) * 16;
  if (n0 >= N || m0 >= M) return;  // wave-uniform exit (EXEC stays full for WMMA)
  const float* Ab = A + (size_t)z * sA;
  const float* Bb = Bm + (size_t)z * sB;
  float* Cb = C + (size_t)z * sC;
  int m = lane & 15, n = lane & 15;
  int koff = (lane >> 4) * 2;
  bool am = (m0 + m) < M;
  bool bn = (n0 + n) < N;
  const float* Arow = Ab + (size_t)(m0 + m) * lda;
  const float* Brow = Bb + (size_t)(n0 + n) * ldb;  // transB path
  v8f acc = {};
#pragma unroll 4
  for (int k = 0; k < K; k += 4) {
    if ((k & 255) == 0 && k + 512 < K) __builtin_prefetch(Arow + k + 512, 0, 0);
    v2f a, bv;
    a.x = am ? Arow[k + koff] : 0.0f;
    a.y = am ? Arow[k + koff + 1] : 0.0f;
    if (transB) {
      bv.x = bn ? Brow[k + koff] : 0.0f;
      bv.y = bn ? Brow[k + koff + 1] : 0.0f;
    } else {
      bv.x = bn ? Bb[(size_t)(k + koff) * ldb + (n0 + n)] : 0.0f;
      bv.y = bn ? Bb[(size_t)(k + koff + 1) * ldb + (n0 + n)] : 0.0f;
    }
    acc = wmma4(a, bv, acc);
  }
#pragma unroll
  for (int r = 0; r < 8; ++r) {
    int mm = m0 + ((lane < 16) ? r : r + 8);
    int nn = n0 + n;
    if (mm < M && nn < N) {
      float v = acc[r] + (bias ? bias[nn] : 0.0f);
      if (relu) v = fmaxf(v, 0.0f);
      Cb[(size_t)mm * ldc + nn] = v;
    }
  }
}

// --------------------------- softmax + per-class decode ----------------------
__global__ __launch_bounds__(128)
void k_stage2(const float* __restrict__ scores, const float* __restrict__ deltas,
              const float* __restrict__ props, float* __restrict__ fgo,
              float* __restrict__ fgr, float* __restrict__ fgs) {
  int roi = blockIdx.x, b = blockIdx.z, t = threadIdx.x;
  __shared__ float red[128];
  __shared__ float prob[80];
  const float* S = scores + ((size_t)b * 1000 + roi) * 80;
  float v = (t < 80) ? S[t] : -1e30f;
  red[t] = v; __syncthreads();
  for (int s = 64; s > 0; s >>= 1) { if (t < s) red[t] = fmaxf(red[t], red[t + s]); __syncthreads(); }
  float mx = red[0]; __syncthreads();
  float e = (t < 80) ? expf(v - mx) : 0.0f;
  red[t] = e; __syncthreads();
  for (int s = 64; s > 0; s >>= 1) { if (t < s) red[t] += red[t + s]; __syncthreads(); }
  float sum = red[0];
  if (t < 80) prob[t] = e / sum;
  __syncthreads();
  if (t >= 1 && t < 80) {
    int cls = t;
    const float* P = props + ((size_t)b * 1000 + roi) * 4;
    float w = P[2] - P[0], h = P[3] - P[1];
    float cx = P[0] + 0.5f * w, cy = P[1] + 0.5f * h;
    const float* D = deltas + ((size_t)b * 1000 + roi) * 320 + cls * 4;
    float dx = D[0] * 0.1f, dy = D[1] * 0.1f;
    float dw = fminf(D[2] * 0.2f, LOGC), dh = fminf(D[3] * 0.2f, LOGC);
    float pcx = dx * w + cx, pcy = dy * h + cy;
    float pw = expf(dw) * w, ph = expf(dh) * h;
    float x1 = clip832(pcx - 0.5f * pw), y1 = clip832(pcy - 0.5f * ph);
    float x2 = clip832(pcx + 0.5f * pw), y2 = clip832(pcy + 0.5f * ph);
    size_t i = (size_t)b * 79000 + (size_t)roi * 79 + (cls - 1);
    fgr[i * 4 + 0] = x1; fgr[i * 4 + 1] = y1; fgr[i * 4 + 2] = x2; fgr[i * 4 + 3] = y2;
    float off = cls * 10000.0f;
    fgo[i * 4 + 0] = x1 + off; fgo[i * 4 + 1] = y1 + off;
    fgo[i * 4 + 2] = x2 + off; fgo[i * 4 + 3] = y2 + off;
    float sc = prob[cls];
    bool ok = (sc > 0.05f) && (x2 - x1 >= 0.01f) && (y2 - y1 >= 0.01f);
    fgs[i] = ok ? sc : NEGV;
  }
}

// --------------------------- final detection write ---------------------------
__global__ __launch_bounds__(128)
void k_det(const float* __restrict__ fgr, const int* __restrict__ idx,
           const float* __restrict__ val, const float* __restrict__ selsc,
           float* __restrict__ out) {
  int b = blockIdx.z, i = threadIdx.x;
  if (i >= 100) return;
  int j = idx[b * 100 + i];
  float v = val[b * 100 + i];
  for (int c = 0; c < 4; ++c)
    out[b * 400 + i * 4 + c] = fgr[((size_t)b * 79000 + j) * 4 + c] * v;
  out[800 + b * 100 + i] = (v > 0.5f) ? selsc[b * 100 + i] : 0.0f;
  out[1000 + b * 100 + i] = (v > 0.5f) ? (float)(j % 79 + 1) : 0.0f;
}

// ===========================================================================
extern "C" void kernel_launch(void* const* d_in, const int* in_sizes, int n_in,
                              void* d_out, int out_size, void* d_ws, size_t ws_size,
                              hipStream_t stream) {
  (void)in_sizes; (void)n_in; (void)out_size; (void)ws_size;
  const float* feat[5];
  for (int i = 0; i < 5; ++i) feat[i] = (const float*)d_in[i];
  const float* lang   = (const float*)d_in[5];
  const float* w_conv = (const float*)d_in[6];
  const float* b_conv = (const float*)d_in[7];
  const float* w_cls  = (const float*)d_in[8];
  const float* b_cls  = (const float*)d_in[9];
  const float* w_bbox = (const float*)d_in[10];
  const float* b_bbox = (const float*)d_in[11];
  const float* w_fc6  = (const float*)d_in[12];
  const float* b_fc6  = (const float*)d_in[13];
  const float* w_fc7  = (const float*)d_in[14];
  const float* b_fc7  = (const float*)d_in[15];
  const float* w_cs   = (const float*)d_in[16];
  const float* b_cs   = (const float*)d_in[17];
  const float* w_bp   = (const float*)d_in[18];
  const float* b_bp   = (const float*)d_in[19];
  const float* w_g1   = (const float*)d_in[20];
  const float* b_g1   = (const float*)d_in[21];
  const float* w_g2   = (const float*)d_in[22];
  const float* b_g2   = (const float*)d_in[23];
  float* out = (float*)d_out;

  const int NB = 2;
  const int HS[5] = {208, 104, 52, 26, 13};
  const float STR[5] = {4.f, 8.f, 16.f, 32.f, 64.f};
  const float SZ[5] = {32.f, 64.f, 128.f, 256.f, 512.f};
  int HW[5], cum[6]; cum[0] = 0;
  for (int l = 0; l < 5; ++l) { HW[l] = HS[l] * HS[l]; cum[l + 1] = cum[l] + HW[l]; }
  const int TOTHW = cum[5];  // 57629
  int KL[5], CB[5], CAND = 0;
  for (int l = 0; l < 5; ++l) { KL[l] = (3 * HW[l] < 1000) ? 3 * HW[l] : 1000; CB[l] = CAND; CAND += KL[l]; }

  // --- workspace bump allocator (~140 MB total) ---
  char* wp = (char*)d_ws;
  auto alloc = [&](size_t n) -> void* { void* p = (void*)wp; wp += (n + 255) & ~(size_t)255; return p; };
  float* obj  = (float*)alloc(sizeof(float) * (size_t)NB * 3 * TOTHW);
  float* dlt  = (float*)alloc(sizeof(float) * (size_t)NB * 12 * TOTHW);
  float* abox = (float*)alloc(sizeof(float) * (size_t)NB * 3 * TOTHW * 4);
  float* ascr = (float*)alloc(sizeof(float) * (size_t)NB * 3 * TOTHW);
  float* tscr = (float*)alloc(sizeof(float) * (size_t)NB * 3 * TOTHW);
  float* coff = (float*)alloc(sizeof(float) * (size_t)NB * CAND * 4);
  float* craw = (float*)alloc(sizeof(float) * (size_t)NB * CAND * 4);
  float* csc  = (float*)alloc(sizeof(float) * (size_t)NB * CAND);
  float* nscr = (float*)alloc(sizeof(float) * (size_t)NB * 79000);
  int*   idx1 = (int*)  alloc(sizeof(int) * NB * 1000);
  float* val1 = (float*)alloc(sizeof(float) * NB * 1000);
  float* ssc1 = (float*)alloc(sizeof(float) * NB * 1000);
  float* props= (float*)alloc(sizeof(float) * NB * 1000 * 4);
  float* xbuf = (float*)alloc(sizeof(float) * (size_t)NB * 1000 * 12544);
  float* f6   = (float*)alloc(sizeof(float) * (size_t)NB * 1000 * 256);
  float* f7   = (float*)alloc(sizeof(float) * (size_t)NB * 1000 * 256);
  float* sc2  = (float*)alloc(sizeof(float) * (size_t)NB * 1000 * 80);
  float* dl2  = (float*)alloc(sizeof(float) * (size_t)NB * 1000 * 320);
  float* fgo  = (float*)alloc(sizeof(float) * (size_t)NB * 79000 * 4);
  float* fgr  = (float*)alloc(sizeof(float) * (size_t)NB * 79000 * 4);
  float* fgs  = (float*)alloc(sizeof(float) * (size_t)NB * 79000);
  int*   idx2 = (int*)  alloc(sizeof(int) * NB * 100);
  float* val2 = (float*)alloc(sizeof(float) * NB * 100);
  float* ssc2 = (float*)alloc(sizeof(float) * NB * 100);
  float* xbuf2= (float*)alloc(sizeof(float) * (size_t)NB * 100 * 12544);
  float* y6   = (float*)alloc(sizeof(float) * (size_t)NB * 100 * 256);
  float* y7   = (float*)alloc(sizeof(float) * (size_t)NB * 100 * 256);
  float* g1o  = (float*)alloc(sizeof(float) * (size_t)NB * 100 * 256);
  float* gf   = (float*)alloc(sizeof(float) * (size_t)NB * 100 * 256);

  // --- RPN: conv3x3 (WMMA) + fused 1x1 heads + decode + per-level topk ---
  for (int l = 0; l < 5; ++l) {
    int hw = HW[l];
    float* objL = obj + (size_t)NB * 3 * cum[l];
    float* dltL = dlt + (size_t)NB * 12 * cum[l];
    float* boxL = abox + (size_t)NB * 3 * cum[l] * 4;
    float* scrL = ascr + (size_t)NB * 3 * cum[l];
    float* tscL = tscr + (size_t)NB * 3 * cum[l];
    k_init_bias<<<dim3((15 * hw + 255) / 256, 1, NB), 256, 0, stream>>>(objL, dltL, b_cls, b_bbox, hw);
    k_conv_rpn<<<dim3((hw + 15) / 16, 2, NB), 256, 0, stream>>>(feat[l], w_conv, b_conv, w_cls, w_bbox, objL, dltL, HS[l], HS[l]);
    k_decode<<<dim3((3 * hw + 255) / 256, 1, NB), 256, 0, stream>>>(objL, dltL, boxL, scrL, HS[l], HS[l], STR[l], SZ[l]);
    k_topk<<<dim3(NB, 1, 1), 1024, 0, stream>>>(boxL, scrL, tscL, coff, craw, csc, 3 * hw, KL[l], (float)l, CB[l], CAND);
  }

  // --- NMS1 -> proposals -> ROI align -> box head ---
  k_nms<<<dim3(NB, 1, 1), 1024, 0, stream>>>(coff, csc, nscr, CAND, 1000, 0.7f, idx1, val1, ssc1);
  k_props<<<dim3(4, 1, NB), 256, 0, stream>>>(craw, idx1, val1, props, CAND);
  k_msroi<<<dim3(1000 * 49, 1, NB), 256, 0, stream>>>(feat[0], feat[1], feat[2], feat[3], props, (size_t)4000, xbuf, (size_t)1000 * 12544);
  k_gemm<<<dim3(63, 2, NB), 256, 0, stream>>>(xbuf, w_fc6, b_fc6, f6, 1000, 256, 12544, 12544, 256, 256, (size_t)1000 * 12544, 0, (size_t)1000 * 256, 0, 1);
  k_gemm<<<dim3(63, 2, NB), 256, 0, stream>>>(f6, w_fc7, b_fc7, f7, 1000, 256, 256, 256, 256, 256, (size_t)1000 * 256, 0, (size_t)1000 * 256, 0, 1);
  k_gemm<<<dim3(63, 1, NB), 256, 0, stream>>>(f7, w_cs, b_cs, sc2, 1000, 80, 256, 256, 80, 80, (size_t)1000 * 256, 0, (size_t)1000 * 80, 0, 0);
  k_gemm<<<dim3(63, 3, NB), 256, 0, stream>>>(f7, w_bp, b_bp, dl2, 1000, 320, 256, 256, 320, 320, (size_t)1000 * 256, 0, (size_t)1000 * 320, 0, 0);

  // --- per-class decode + softmax -> NMS2 -> detections ---
  k_stage2<<<dim3(1000, 1, NB), 128, 0, stream>>>(sc2, dl2, props, fgo, fgr, fgs);
  k_nms<<<dim3(NB, 1, 1), 1024, 0, stream>>>(fgo, fgs, nscr, 79000, 100, 0.5f, idx2, val2, ssc2);
  k_det<<<dim3(1, 1, NB), 128, 0, stream>>>(fgr, idx2, val2, ssc2, out);

  // --- grounding branch on the 100 detections ---
  k_msroi<<<dim3(100 * 49, 1, NB), 256, 0, stream>>>(feat[0], feat[1], feat[2], feat[3], out, (size_t)400, xbuf2, (size_t)100 * 12544);
  k_gemm<<<dim3(7, 2, NB), 256, 0, stream>>>(xbuf2, w_fc6, b_fc6, y6, 100, 256, 12544, 12544, 256, 256, (size_t)100 * 12544, 0, (size_t)100 * 256, 0, 1);
  k_gemm<<<dim3(7, 2, NB), 256, 0, stream>>>(y6, w_fc7, b_fc7, y7, 100, 256, 256, 256, 256, 256, (size_t)100 * 256, 0, (size_t)100 * 256, 0, 1);
  k_gemm<<<dim3(7, 2, NB), 256, 0, stream>>>(y7, w_g1, b_g1, g1o, 100, 256, 256, 256, 256, 256, (size_t)100 * 256, 0, (size_t)100 * 256, 0, 1);
  k_gemm<<<dim3(7, 2, NB), 256, 0, stream>>>(g1o, w_g2, b_g2, gf, 100, 256, 256, 256, 256, 256, (size_t)100 * 256, 0, (size_t)100 * 256, 0, 0);
  // gscores = gfeat @ lang_i^T  (transB)
  k_gemm<<<dim3(7, 2, NB), 256, 0, stream>>>(gf, lang, nullptr, out + 1200, 100, 256, 256, 256, 256, 256, (size_t)100 * 256, (size_t)256 * 256, (size_t)100 * 256, 1, 0);
}